// FrameWeightComputer_29197187678412
// MI455X (gfx1250) — compile-verified
//
#include <hip/hip_runtime.h>
#include <stdint.h>

// ---------------------------------------------------------------------------
// FrameWeightComputer for MI455X (gfx1250).
// Memory-bound (0.56 FLOP/B, 536 MB streamed once): optimize the data path.
//  - neighbor_dirs staged global->LDS via CDNA5 async DMA (ASYNCcnt), with
//    th:TH_LOAD_NT so the one-pass stream does not pollute the 192MB L2.
//  - IOFFSET is applied to BOTH the LDS and global side of the async copy
//    (CDNA5 ISA), so each thread needs only ONE global base
//    VGPR pair + ONE LDS base VGPR; the 12 chunks use offset:0..176.
//  - Per-lane-owned chunks: each thread copies and consumes its own frame,
//    so s_wait_asynccnt 0 is the only sync needed (no workgroup barrier).
//  - LDS frame stride padded 48 -> 52 dwords (208B, 16B aligned) to cut
//    ds_load_b128 bank conflicts from 8-way to 2-way.
//  - cur_dir loads and weight stores are non-temporal (no reuse).
// ---------------------------------------------------------------------------

typedef float v4f __attribute__((ext_vector_type(4)));

#define FRAMES_PER_BLOCK 256
#define CHUNKS_PER_FRAME 12          // 48 floats = 12 x 16B chunks
#define PAD_DW           52          // 48 data dwords + 4 pad dwords
#define LDS_DW           (FRAMES_PER_BLOCK * PAD_DW)   // 13312 dw = 53,248 B

// One async 16B copy; OFF is a compile-time byte offset applied to both the
// LDS destination and the global source by the hardware.
#define ASYNC_CP(OFF)                                                          \
    asm volatile("global_load_async_to_lds_b128 %0, %1, off offset:" #OFF      \
                 " th:TH_LOAD_NT"                                              \
                 :: "v"(ldsOff), "v"(gaddr) : "memory")

__global__ __launch_bounds__(FRAMES_PER_BLOCK)
void FrameWeightComputer_29197187678412_kernel(const float* __restrict__ cur,
                                               const float* __restrict__ nb,
                                               float* __restrict__ out,
                                               int B)
{
    __shared__ float lds[LDS_DW];

    const int t     = threadIdx.x;
    const int frame = blockIdx.x * FRAMES_PER_BLOCK + t;

    // Base addresses: one VGPR pair (global) + one VGPR (LDS) per thread.
    const uint32_t ldsOff = (uint32_t)(uintptr_t)&lds[0] + (uint32_t)(t * (PAD_DW * 4));
    const void*    gaddr  = (const void*)(nb + (size_t)frame * 48);

    if (frame < B) {
        ASYNC_CP(0);   ASYNC_CP(16);  ASYNC_CP(32);  ASYNC_CP(48);
        ASYNC_CP(64);  ASYNC_CP(80);  ASYNC_CP(96);  ASYNC_CP(112);
        ASYNC_CP(128); ASYNC_CP(144); ASYNC_CP(160); ASYNC_CP(176);
    }
    // Drain this wave's async copies; each thread reads only its own data,
    // so no workgroup barrier is required.
    asm volatile("s_wait_asynccnt 0" ::: "memory");

    if (frame >= B) return;

    // ---- Pull this thread's frame (48 floats) out of LDS -------------------
    const v4f* lp = (const v4f*)&lds[t * PAD_DW];
    float v[48];
    #pragma unroll
    for (int i = 0; i < CHUNKS_PER_FRAME; ++i) {
        v4f q = lp[i];
        v[4 * i + 0] = q.x;
        v[4 * i + 1] = q.y;
        v[4 * i + 2] = q.z;
        v[4 * i + 3] = q.w;
    }

    // cur_dir: 3 floats, streamed once -> non-temporal loads.
    const float* cp = cur + (size_t)frame * 3;
    const float cx = __builtin_nontemporal_load(cp + 0);
    const float cy = __builtin_nontemporal_load(cp + 1);
    const float cz = __builtin_nontemporal_load(cp + 2);

    // ---- dots -> exp(dot - 1) ----------------------------------------------
    float e[16];
    #pragma unroll
    for (int n = 0; n < 16; ++n) {
        const float d = fmaf(v[3 * n + 0], cx,
                        fmaf(v[3 * n + 1], cy,
                             v[3 * n + 2] * cz));
        e[n] = __expf(d - 1.0f);              // S_WEIGHT == 1
    }

    // relu(e - min(e)) == e - min(e) (min element -> exactly 0)
    float mn = e[0];
    #pragma unroll
    for (int n = 1; n < 16; ++n) mn = fminf(mn, e[n]);

    float s = 0.0f;
    #pragma unroll
    for (int n = 0; n < 16; ++n) s += (e[n] - mn);

    const float inv = 1.0f / s;               // one accurate divide, 16 muls

    // ---- 4 x 16B non-temporal stores (64B/frame, contiguous per wave) ------
    v4f* op = (v4f*)(out + (size_t)frame * 16);
    #pragma unroll
    for (int g = 0; g < 4; ++g) {
        v4f o;
        o.x = (e[4 * g + 0] - mn) * inv;
        o.y = (e[4 * g + 1] - mn) * inv;
        o.z = (e[4 * g + 2] - mn) * inv;
        o.w = (e[4 * g + 3] - mn) * inv;
        __builtin_nontemporal_store(o, op + g);
    }
}

extern "C" void kernel_launch(void* const* d_in, const int* in_sizes, int n_in,
                              void* d_out, int out_size, void* d_ws, size_t ws_size,
                              hipStream_t stream) {
    const float* cur = (const float*)d_in[0];   // [B, 3]
    const float* nb  = (const float*)d_in[1];   // [B, 16, 3]
    float*       out = (float*)d_out;           // [B, 16]

    const int B    = in_sizes[0] / 3;
    const int grid = (B + FRAMES_PER_BLOCK - 1) / FRAMES_PER_BLOCK;

    hipLaunchKernelGGL(FrameWeightComputer_29197187678412_kernel,
                       dim3(grid), dim3(FRAMES_PER_BLOCK), 0, stream,
                       cur, nb, out, B);
}